// CrossRMSD_40819369181337
// MI455X (gfx1250) — compile-verified
//
#include <hip/hip_runtime.h>

typedef __attribute__((ext_vector_type(2))) float v2f;
typedef __attribute__((ext_vector_type(8))) float v8f;

#define S_COUNT 2048
#define T_COUNT 2048
#define NATOMS  128

// ---------------------------------------------------------------------------
// Prep: center one structure, emit planar (coordinate-major) matrices + norms.
// planes[i*S_COUNT*NATOMS + s*NATOMS + a] = X[s,a,i] - mean_i(s)
// ---------------------------------------------------------------------------
__global__ __launch_bounds__(128)
void crmsd_center(const float* __restrict__ X, float* __restrict__ planes,
                  float* __restrict__ norms)
{
    __shared__ float sred[128];
    const int s = blockIdx.x;
    const int a = threadIdx.x;

    const float* xp = X + ((size_t)s * NATOMS + a) * 3;
    float x = xp[0], y = xp[1], z = xp[2];

    float coord[3] = {x, y, z};
    float mean[3];
#pragma unroll
    for (int c = 0; c < 3; ++c) {
        sred[a] = coord[c];
        __syncthreads();
        for (int off = 64; off > 0; off >>= 1) {
            if (a < off) sred[a] += sred[a + off];
            __syncthreads();
        }
        mean[c] = sred[0] * (1.0f / NATOMS);
        __syncthreads();
    }

    const float cx = x - mean[0], cy = y - mean[1], cz = z - mean[2];

    sred[a] = cx * cx + cy * cy + cz * cz;
    __syncthreads();
    for (int off = 64; off > 0; off >>= 1) {
        if (a < off) sred[a] += sred[a + off];
        __syncthreads();
    }
    if (a == 0) norms[s] = sred[0];

    planes[((size_t)(0 * S_COUNT + s)) * NATOMS + a] = cx;
    planes[((size_t)(1 * S_COUNT + s)) * NATOMS + a] = cy;
    planes[((size_t)(2 * S_COUNT + s)) * NATOMS + a] = cz;
}

// ---------------------------------------------------------------------------
// Main: one wave computes a 16x16 tile of (s,t) pairs.
//  Phase 1 (WMMA): 9 f32 accumulators (3x3 covariance components), K=128 via
//    32 x v_wmma_f32_16x16x4_f32 per component, fragments streamed as one
//    global_load_b64 per lane per plane per k-step (A 16x4 f32 layout:
//    lane -> row M = lane&15, K = k0 + 2*(lane>>4)).
//  Phase 2 (VALU): C/D layout puts pair (M = v + 8*(lane>>4), N = lane&15) in
//    VGPR v of lane, so each lane owns 8 pairs with all 9 R entries in
//    registers. F (4x4 symmetric) coefficients are extracted for all 8 pairs
//    (freeing the accumulators), then the 50 power iterations run with FOUR
//    independent pair-chains interleaved to hide the sqrt/rcp/fma latency
//    chain inside a single wave.
// ---------------------------------------------------------------------------
__global__ __launch_bounds__(256)
void crmsd_main(const float* __restrict__ Apl, const float* __restrict__ Bpl,
                const float* __restrict__ normM, const float* __restrict__ normT,
                float* __restrict__ out)
{
    const int lane  = threadIdx.x & 31;
    const int wave  = threadIdx.x >> 5;
    const int sBase = blockIdx.y << 4;
    const int tBase = ((blockIdx.x << 3) + wave) << 4;

    const int mrow = lane & 15;
    const int koff = (lane >> 4) << 1;   // lanes 16..31 take K+2, K+3

    const float* aP[3];
    const float* bP[3];
#pragma unroll
    for (int i = 0; i < 3; ++i) {
        aP[i] = Apl + ((size_t)(i * S_COUNT + sBase + mrow)) * NATOMS + koff;
        bP[i] = Bpl + ((size_t)(i * T_COUNT + tBase + mrow)) * NATOMS + koff;
    }

    v8f c[3][3];
#pragma unroll
    for (int i = 0; i < 3; ++i)
#pragma unroll
        for (int j = 0; j < 3; ++j)
            c[i][j] = (v8f)0.0f;

#pragma unroll 4
    for (int k0 = 0; k0 < NATOMS; k0 += 4) {
        v2f af[3], bf[3];
#pragma unroll
        for (int i = 0; i < 3; ++i) {
            af[i] = *(const v2f*)(aP[i] + k0);
            bf[i] = *(const v2f*)(bP[i] + k0);
        }
#pragma unroll
        for (int i = 0; i < 3; ++i)
#pragma unroll
            for (int j = 0; j < 3; ++j)
                c[i][j] = __builtin_amdgcn_wmma_f32_16x16x4_f32(
                    /*neg_a=*/false, af[i], /*neg_b=*/false, bf[j],
                    /*c_mod=*/(short)0, c[i][j],
                    /*reuse_a=*/false, /*reuse_b=*/false);
    }

    // ---- Extract 4x4 symmetric F per pair (indices: 0:00 1:01 2:02 3:03
    //      4:11 5:12 6:13 7:22 8:23 9:33). Matches reference R_to_F map. ----
    float F[8][10];
#pragma unroll
    for (int v = 0; v < 8; ++v) {
        const float R00 = c[0][0][v], R01 = c[0][1][v], R02 = c[0][2][v];
        const float R10 = c[1][0][v], R11 = c[1][1][v], R12 = c[1][2][v];
        const float R20 = c[2][0][v], R21 = c[2][1][v], R22 = c[2][2][v];
        F[v][0] =  R00 + R11 + R22;
        F[v][1] =  R12 - R21;
        F[v][2] =  R20 - R02;
        F[v][3] =  R01 - R10;
        F[v][4] =  R00 - R11 - R22;
        F[v][5] =  R01 + R10;
        F[v][6] =  R02 + R20;
        F[v][7] = -R00 + R11 - R22;
        F[v][8] =  R12 + R21;
        F[v][9] = -R00 - R11 + R22;
    }

    const int   tcol    = tBase + (lane & 15);
    const float nT      = normT[tcol];
    const float inv_den = 1.0f / ((float)NATOMS + 1e-5f);

    // ---- Power iteration: two groups of 4 interleaved independent chains ----
#pragma unroll
    for (int g = 0; g < 2; ++g) {
        float G0[4], G1[4], G2[4], G3[4];
        float q[4][4];
#pragma unroll
        for (int u = 0; u < 4; ++u) {
            const int p = 4 * g + u;
            // shift = ||F||_F (upper bound on spectral radius)
            const float fro2 =
                F[p][0]*F[p][0] + F[p][4]*F[p][4] +
                F[p][7]*F[p][7] + F[p][9]*F[p][9] +
                2.0f * (F[p][1]*F[p][1] + F[p][2]*F[p][2] + F[p][3]*F[p][3] +
                        F[p][5]*F[p][5] + F[p][6]*F[p][6] + F[p][8]*F[p][8]);
            const float shift = sqrtf(fro2);
            G0[u] = F[p][0] + shift;
            G1[u] = F[p][4] + shift;
            G2[u] = F[p][7] + shift;
            G3[u] = F[p][9] + shift;
            q[u][0] = 0.5f; q[u][1] = 0.5f; q[u][2] = 0.5f; q[u][3] = 0.5f;
        }

#pragma unroll 1
        for (int it = 0; it < 50; ++it) {
#pragma unroll
            for (int u = 0; u < 4; ++u) {
                const int p = 4 * g + u;
                const float w0 = G0[u]  *q[u][0] + F[p][1]*q[u][1] +
                                 F[p][2]*q[u][2] + F[p][3]*q[u][3];
                const float w1 = F[p][1]*q[u][0] + G1[u]  *q[u][1] +
                                 F[p][5]*q[u][2] + F[p][6]*q[u][3];
                const float w2 = F[p][2]*q[u][0] + F[p][5]*q[u][1] +
                                 G2[u]  *q[u][2] + F[p][8]*q[u][3];
                const float w3 = F[p][3]*q[u][0] + F[p][6]*q[u][1] +
                                 F[p][8]*q[u][2] + G3[u]  *q[u][3];
                const float n   = sqrtf(w0*w0 + w1*w1 + w2*w2 + w3*w3);
                const float inv = 1.0f / (n + 1e-12f);
                q[u][0] = w0 * inv; q[u][1] = w1 * inv;
                q[u][2] = w2 * inv; q[u][3] = w3 * inv;
            }
        }

#pragma unroll
        for (int u = 0; u < 4; ++u) {
            const int p = 4 * g + u;
            // Rayleigh quotient with UNshifted F (q is unit norm)
            const float w0 = F[p][0]*q[u][0] + F[p][1]*q[u][1] +
                             F[p][2]*q[u][2] + F[p][3]*q[u][3];
            const float w1 = F[p][1]*q[u][0] + F[p][4]*q[u][1] +
                             F[p][5]*q[u][2] + F[p][6]*q[u][3];
            const float w2 = F[p][2]*q[u][0] + F[p][5]*q[u][1] +
                             F[p][7]*q[u][2] + F[p][8]*q[u][3];
            const float w3 = F[p][3]*q[u][0] + F[p][6]*q[u][1] +
                             F[p][8]*q[u][2] + F[p][9]*q[u][3];
            const float eig = q[u][0]*w0 + q[u][1]*w1 + q[u][2]*w2 + q[u][3]*w3;

            const int srow = sBase + p + ((lane >> 4) << 3);
            const float nM = normM[srow];
            float sq = (nM + nT - 2.0f * eig) * inv_den;
            sq = sq > 0.0f ? sq : 0.0f;
            out[(size_t)srow * T_COUNT + tcol] = sqrtf(sq);
        }
    }
}

// ---------------------------------------------------------------------------
extern "C" void kernel_launch(void* const* d_in, const int* in_sizes, int n_in,
                              void* d_out, int out_size, void* d_ws, size_t ws_size,
                              hipStream_t stream) {
    (void)in_sizes; (void)n_in; (void)out_size; (void)ws_size;

    const float* Xm = (const float*)d_in[0];   // (2048,128,3) f32
    const float* Xt = (const float*)d_in[1];   // (2048,128,3) f32
    float* out = (float*)d_out;                // (2048,2048) f32

    float* ws   = (float*)d_ws;
    float* Apl  = ws;                                   // 3*2048*128 floats
    float* Bpl  = Apl + (size_t)3 * S_COUNT * NATOMS;   // 3*2048*128 floats
    float* nM   = Bpl + (size_t)3 * T_COUNT * NATOMS;   // 2048 floats
    float* nT   = nM + S_COUNT;                         // 2048 floats

    crmsd_center<<<S_COUNT, 128, 0, stream>>>(Xm, Apl, nM);
    crmsd_center<<<T_COUNT, 128, 0, stream>>>(Xt, Bpl, nT);

    dim3 grid(T_COUNT / (16 * 8), S_COUNT / 16);   // (16, 128)
    crmsd_main<<<grid, 256, 0, stream>>>(Apl, Bpl, nM, nT, out);
}